// IndirectAccessKVCacheAttention_58265526338179
// MI455X (gfx1250) — compile-verified
//
#include <hip/hip_runtime.h>
#include <math.h>

#define MAX_SEQ 2048
#define BATCH   16
#define HEADS   32
#define DMODEL  128
#define OFFSET  2047
#define TFULL   2048   // OFFSET + 1 (history + new token)

typedef float v2f __attribute__((ext_vector_type(2)));
typedef float v8f __attribute__((ext_vector_type(8)));

__device__ __forceinline__ float readlane_f32(float v, int srcLane) {
    return __uint_as_float(__builtin_amdgcn_readlane(__float_as_uint(v), srcLane));
}

// ---------------------------------------------------------------------------
// Kernel 1: beam-trace suffix scan.
// trace[t] = f_t o f_{t+1} o ... o f_{OFFSET-1}  (f_t(x) = beam_idx[t][x])
// 64 chunk compositions (32 steps each, 16 lanes per chunk), serial suffix
// combine of the 64 chunk maps in LDS, then parallel re-walk.
// ---------------------------------------------------------------------------
__global__ __launch_bounds__(1024)
void beam_trace_kernel(const int* __restrict__ beam_idx, int* __restrict__ trace)
{
    __shared__ int S[64][16];   // per-chunk composition
    __shared__ int R[64][16];   // suffix composition of chunks after g
    const int tid = threadIdx.x;
    const int g   = tid >> 4;   // chunk id 0..63
    const int x   = tid & 15;   // beam slot
    const int CH  = 32;
    const int t0  = g * CH;
    const int t1  = (t0 + CH < OFFSET) ? (t0 + CH) : OFFSET;

    // phase 1: S_g = f_{t0} o ... o f_{t1-1}
    int cur = x;
    for (int t = t1 - 1; t >= t0; --t)
        cur = beam_idx[t * BATCH + cur];
    S[g][x] = cur;
    __syncthreads();

    // phase 2: R_g = S_{g+1} o ... o S_{63}   (16 threads, LDS only)
    if (tid < 16) {
        int c = x;
        R[63][x] = c;
        for (int gg = 62; gg >= 0; --gg) {
            c = S[gg + 1][c];
            R[gg][x] = c;
        }
    }
    __syncthreads();

    // phase 3: re-walk chunk starting from R_g, emit trace
    cur = R[g][x];
    for (int t = t1 - 1; t >= t0; --t) {
        cur = beam_idx[t * BATCH + cur];
        trace[t * BATCH + x] = cur;
    }
}

// ---------------------------------------------------------------------------
// Kernel 2: flash-decode attention, one workgroup per (b,h).
// 8 waves x 256 t each; scores via V_WMMA_F32_16X16X4_F32 (16 t per group,
// K=4 chained over 32 d-chunks); online softmax with v_readlane broadcasts;
// coalesced float4 V loads off wave-uniform (SGPR) row bases.
// ---------------------------------------------------------------------------
__global__ __launch_bounds__(256)
void ia_kv_attn_kernel(const float* __restrict__ q,
                       const float* __restrict__ knew,
                       const float* __restrict__ vnew,
                       const float* __restrict__ kcache,
                       const float* __restrict__ vcache,
                       const int*   __restrict__ trace,
                       const float* __restrict__ mask,
                       float*       __restrict__ out)
{
    const int h    = blockIdx.x;
    const int b    = blockIdx.y;
    const int tid  = threadIdx.x;
    const int lane = tid & 31;
    const int wave = tid >> 5;

    __shared__ __align__(16) float qs[DMODEL];
    __shared__ float red_m[8];
    __shared__ float red_l[8];
    __shared__ __align__(16) float red_acc[8][DMODEL];

    const size_t bh = (size_t)b * HEADS + h;
    if (tid < DMODEL) qs[tid] = q[bh * DMODEL + tid];
    __syncthreads();

    const float rscale = 0.088388347648318447f;   // 1/sqrt(128)
    const int   hh2    = (lane >> 4) << 1;        // K sub-offset: 0 or 2

    float  mcur = -3.402823466e38f;
    float  lsum = 0.f;
    float4 acc  = make_float4(0.f, 0.f, 0.f, 0.f);

    const int tbeg = wave * (TFULL / 8);
    const int tend = tbeg + (TFULL / 8);

    for (int t0 = tbeg; t0 < tend; t0 += 16) {
        // one K-row id per lane (lanes 16-31 mirror 0-15); -1 => new token
        const int tt = t0 + (lane & 15);
        int rowid = -1;
        if (tt < OFFSET) rowid = tt * BATCH + trace[tt * BATCH + b];

        const float* krow = (rowid >= 0)
            ? (kcache + ((size_t)rowid * HEADS + h) * DMODEL)
            : (knew + bh * DMODEL);
        const float* vrow_pf = (rowid >= 0)
            ? (vcache + ((size_t)rowid * HEADS + h) * DMODEL)
            : (vnew + bh * DMODEL);
        __builtin_prefetch(vrow_pf, 0, 0);   // global_prefetch_b8: warm V row

        // ---- scores for 16 t via chained f32 WMMA ----
        // A: lane (m = lane&15) holds k_{t0+m}[4c + hh2 .. +1]  (16x4 tile)
        // B: q[4c + hh2 .. +1] broadcast across all 16 columns  (4x16 tile)
        v8f C = {0.f, 0.f, 0.f, 0.f, 0.f, 0.f, 0.f, 0.f};
#pragma unroll 8
        for (int c = 0; c < 32; ++c) {
            v2f a  = *(const v2f*)(krow + 4 * c + hh2);
            v2f bq = *(const v2f*)(&qs[4 * c + hh2]);
            C = __builtin_amdgcn_wmma_f32_16x16x4_f32(
                    false, a, false, bq, (short)0, C, false, false);
        }

        // extract the 16 scores as wave-uniform scalars via v_readlane
        // (C vgpr r: lanes0-15 -> M=r, lanes16-31 -> M=r+8)
        float s[16];
#pragma unroll
        for (int m = 0; m < 16; ++m) {
            float sv = readlane_f32(C[m & 7], (m >> 3) * 16);
            s[m] = sv * rscale + mask[(size_t)b * TFULL + (t0 + m)];
        }

        // ---- online softmax update + PV accumulation ----
        float mnew = mcur;
#pragma unroll
        for (int m = 0; m < 16; ++m) mnew = fmaxf(mnew, s[m]);
        const float corr = __expf(mcur - mnew);
        lsum *= corr;
        acc.x *= corr; acc.y *= corr; acc.z *= corr; acc.w *= corr;

#pragma unroll
        for (int m = 0; m < 16; ++m) {
            const float p = __expf(s[m] - mnew);
            lsum += p;
            // wave-uniform row id -> SGPR base address for the V row
            const int r = __builtin_amdgcn_readlane(rowid, m);
            const float* vrow = (r >= 0)
                ? (vcache + ((size_t)r * HEADS + h) * DMODEL)
                : (vnew + bh * DMODEL);
            const float4 v4 = *(const float4*)(vrow + 4 * lane);  // coalesced 512B
            acc.x += p * v4.x; acc.y += p * v4.y;
            acc.z += p * v4.z; acc.w += p * v4.w;
        }
        mcur = mnew;
    }

    // ---- cross-wave combine ----
    if (lane == 0) { red_m[wave] = mcur; red_l[wave] = lsum; }
    *(float4*)&red_acc[wave][4 * lane] = acc;
    __syncthreads();

    if (wave == 0) {
        float M = red_m[0];
#pragma unroll
        for (int w = 1; w < 8; ++w) M = fmaxf(M, red_m[w]);
        float  L = 0.f;
        float4 o = make_float4(0.f, 0.f, 0.f, 0.f);
#pragma unroll
        for (int w = 0; w < 8; ++w) {
            const float  cw = __expf(red_m[w] - M);
            const float4 aw = *(const float4*)&red_acc[w][4 * lane];
            L   += red_l[w] * cw;
            o.x += cw * aw.x; o.y += cw * aw.y;
            o.z += cw * aw.z; o.w += cw * aw.w;
        }
        const float invL = 1.0f / L;
        o.x *= invL; o.y *= invL; o.z *= invL; o.w *= invL;
        *(float4*)&out[bh * DMODEL + 4 * lane] = o;
    }
}

// ---------------------------------------------------------------------------
extern "C" void kernel_launch(void* const* d_in, const int* in_sizes, int n_in,
                              void* d_out, int out_size, void* d_ws, size_t ws_size,
                              hipStream_t stream)
{
    const float* q      = (const float*)d_in[0];   // (B,1,H,D)
    const float* knew   = (const float*)d_in[1];   // (B,1,H,D)
    const float* vnew   = (const float*)d_in[2];   // (B,1,H,D)
    const float* kcache = (const float*)d_in[3];   // (T,B,H,D)
    const float* vcache = (const float*)d_in[4];   // (T,B,H,D)
    const int*   beam   = (const int*)d_in[5];     // (T,B)
    const float* mask   = (const float*)d_in[6];   // (B,1,1,T)
    (void)in_sizes; (void)n_in; (void)out_size; (void)ws_size;

    int* trace = (int*)d_ws;                       // OFFSET*BATCH ints

    beam_trace_kernel<<<1, 1024, 0, stream>>>(beam, trace);

    dim3 grid(HEADS, BATCH);
    ia_kv_attn_kernel<<<grid, 256, 0, stream>>>(
        q, knew, vnew, kcache, vcache, trace, mask, (float*)d_out);
}